// InteractionNetwork_86088324481849
// MI455X (gfx1250) — compile-verified
//
#include <hip/hip_runtime.h>
#include <stdint.h>

typedef __attribute__((ext_vector_type(2))) float v2f;
typedef __attribute__((ext_vector_type(8))) float v8f;

static constexpr int   DIM    = 128;        // feature dim
static constexpr int   MT     = 64;         // rows per block (4 x 16 sub-tiles)
static constexpr int   RT     = MT / 16;    // row sub-tiles per wave
static constexpr int   XS_E   = 3 * DIM + 4; // edge X row stride (bank-pad)
static constexpr int   XS_N   = 2 * DIM + 4; // node X row stride
static constexpr int   HS     = DIM + 4;     // hidden row stride
static constexpr float LN_EPS = 1e-5f;

// ---------------------------------------------------------------------------
// CDNA5 async global->LDS copy (ASYNCcnt-tracked), GV addressing mode.
// LDS destination offset = low 32 bits of the generic shared-memory address.
// ---------------------------------------------------------------------------
__device__ __forceinline__ void async_copy_b128(void* lds_dst, const void* gsrc)
{
  unsigned lds = (unsigned)(uintptr_t)lds_dst;
  asm volatile("global_load_async_to_lds_b128 %0, %1, off"
               :: "v"(lds), "v"(gsrc) : "memory");
}
__device__ __forceinline__ void async_wait0()
{
  asm volatile("s_wait_asynccnt 0x0" ::: "memory");
}

// ---------------------------------------------------------------------------
// RT independent 16x16 output tiles of X[MT,K] @ W[K,128] + bias, one wave.
// A frag (16x4 f32): lanes 0-15 carry K=0,1; lanes 16-31 carry K=2,3.
// One B fragment is shared by all RT row sub-tiles (4x L2-traffic reuse) and
// the RT accumulators form independent WMMA chains (latency hiding).
// ---------------------------------------------------------------------------
template <int K>
__device__ __forceinline__ void gemm_tiles_f32(const float* Xl, int xs,
                                               const float* __restrict__ W,
                                               int lane, int nc, float bias,
                                               v8f acc[RT])
{
  const int m  = lane & 15;
  const int kh = (lane >> 4) << 1;   // 0 or 2
#pragma unroll
  for (int r = 0; r < RT; ++r)
#pragma unroll
    for (int i = 0; i < 8; ++i) acc[r][i] = bias;

#pragma unroll 2
  for (int k = 0; k < K; k += 4) {
    v2f b;
    b.x = W[(size_t)(k + kh) * DIM + nc];
    b.y = W[(size_t)(k + kh + 1) * DIM + nc];
#pragma unroll
    for (int r = 0; r < RT; ++r) {
      v2f a;
      a.x = Xl[(r * 16 + m) * xs + k + kh];
      a.y = Xl[(r * 16 + m) * xs + k + kh + 1];
      acc[r] = __builtin_amdgcn_wmma_f32_16x16x4_f32(false, a, false, b,
                                                     (short)0, acc[r],
                                                     false, false);
    }
  }
}

// Store RT 16x16 f32 C/D tiles to LDS (row stride hs floats).
// C/D layout: VGPR p holds M = p + 8*(lane/16), N = nc.
__device__ __forceinline__ void store_tiles(float* Hl, int hs, int lane, int nc,
                                            const v8f acc[RT], bool relu)
{
  const int mb = (lane >> 4) << 3;
#pragma unroll
  for (int r = 0; r < RT; ++r)
#pragma unroll
    for (int i = 0; i < 8; ++i) {
      float x = acc[r][i];
      if (relu && x < 0.f) x = 0.f;
      Hl[(r * 16 + mb + i) * hs + nc] = x;
    }
}

// ---------------------------------------------------------------------------
__global__ void __launch_bounds__(256) zero_f32(float* __restrict__ p, int n)
{
  for (int i = blockIdx.x * 256 + threadIdx.x; i < n; i += gridDim.x * 256)
    p[i] = 0.f;
}

// ---------------------------------------------------------------------------
// Edge MLP: 64 edges per block. X = [edge_attr | feat[src] | feat[dst]]
// -> relu(XW1+b1) -> relu(·W2+b2) -> ·W3+b3 -> LayerNorm.
// Writes edge output; atomically scatter-adds into aggr[receiver].
// ---------------------------------------------------------------------------
__global__ void __launch_bounds__(256, 1) edge_mlp(
    const float* __restrict__ feat, const int* __restrict__ eidx,
    const float* __restrict__ eattr,
    const float* __restrict__ W1, const float* __restrict__ b1,
    const float* __restrict__ W2, const float* __restrict__ b2,
    const float* __restrict__ W3, const float* __restrict__ b3,
    const float* __restrict__ gamma, const float* __restrict__ beta,
    float* __restrict__ eout, float* __restrict__ aggr, int E)
{
  extern __shared__ float smem[];
  float* X  = smem;                 // [MT][XS_E]
  float* Ha = X + MT * XS_E;        // [MT][HS]
  float* Hb = Ha + MT * HS;         // [MT][HS]
  int*   sr = (int*)(Hb + MT * HS);
  int*   rr = sr + MT;

  const int t  = threadIdx.x;
  const int r0 = blockIdx.x * MT;

  if (t < MT) {
    int e = r0 + t; if (e >= E) e = E - 1;
    sr[t] = eidx[e];
    rr[t] = eidx[E + e];
  }
  __syncthreads();

  // Async gather [edge_attr | feat[s] | feat[r]] -> X : MT rows x 96 float4
#pragma unroll
  for (int i = 0; i < MT * 96 / 256; ++i) {
    int idx = t + i * 256;
    int row = idx / 96;
    int col = (idx % 96) * 4;
    int e   = r0 + row; if (e >= E) e = E - 1;
    const float* src;
    if (col < DIM)          src = eattr + (size_t)e * DIM + col;
    else if (col < 2 * DIM) src = feat + (size_t)sr[row] * DIM + (col - DIM);
    else                    src = feat + (size_t)rr[row] * DIM + (col - 2 * DIM);
    async_copy_b128(&X[row * XS_E + col], src);
  }
  async_wait0();
  __syncthreads();

  const int wave = t >> 5, lane = t & 31;
  const int nc   = wave * 16 + (lane & 15);   // this wave's output column

  v8f acc[RT];
  gemm_tiles_f32<3 * DIM>(X, XS_E, W1, lane, nc, b1[nc], acc);
  store_tiles(Ha, HS, lane, nc, acc, true);
  __syncthreads();
  gemm_tiles_f32<DIM>(Ha, HS, W2, lane, nc, b2[nc], acc);
  store_tiles(Hb, HS, lane, nc, acc, true);
  __syncthreads();
  gemm_tiles_f32<DIM>(Hb, HS, W3, lane, nc, b3[nc], acc);
  store_tiles(Ha, HS, lane, nc, acc, false);
  __syncthreads();

  // LayerNorm: 16 threads per row, 8 cols each; MT/16 passes over 256 threads.
#pragma unroll
  for (int pass = 0; pass < MT / 16; ++pass) {
    const int row = (t >> 4) + pass * 16;
    const int sub = t & 15;
    float s = 0.f, ss = 0.f;
#pragma unroll
    for (int j = 0; j < 8; ++j) {
      float v = Ha[row * HS + sub * 8 + j];
      s += v; ss += v * v;
    }
#pragma unroll
    for (int off = 1; off < 16; off <<= 1) {  // half-wave reduction (wave32)
      s  += __shfl_xor(s,  off, 16);
      ss += __shfl_xor(ss, off, 16);
    }
    const float mu  = s * (1.f / DIM);
    const float inv = rsqrtf(ss * (1.f / DIM) - mu * mu + LN_EPS);
    const int e  = r0 + row;
    const int rv = rr[row];
    if (e < E) {
#pragma unroll
      for (int j = 0; j < 8; ++j) {
        int c = sub * 8 + j;
        float v = (Ha[row * HS + c] - mu) * inv * gamma[c] + beta[c];
        eout[(size_t)e * DIM + c] = v;
        atomicAdd(&aggr[(size_t)rv * DIM + c], v);
      }
    }
  }
}

// ---------------------------------------------------------------------------
// Node MLP: 64 nodes per block. X = [feat | aggr] -> MLP -> LayerNorm.
// ---------------------------------------------------------------------------
__global__ void __launch_bounds__(256, 1) node_mlp(
    const float* __restrict__ feat, const float* __restrict__ aggr,
    const float* __restrict__ W1, const float* __restrict__ b1,
    const float* __restrict__ W2, const float* __restrict__ b2,
    const float* __restrict__ W3, const float* __restrict__ b3,
    const float* __restrict__ gamma, const float* __restrict__ beta,
    float* __restrict__ nout, int N)
{
  extern __shared__ float smem[];
  float* X  = smem;                 // [MT][XS_N]
  float* Ha = X + MT * XS_N;        // [MT][HS]
  float* Hb = Ha + MT * HS;         // [MT][HS]

  const int t  = threadIdx.x;
  const int r0 = blockIdx.x * MT;

  // Async gather [feat | aggr] -> X : MT rows x 64 float4
#pragma unroll
  for (int i = 0; i < MT * 64 / 256; ++i) {
    int idx = t + i * 256;
    int row = idx >> 6;
    int col = (idx & 63) * 4;
    int n   = r0 + row; if (n >= N) n = N - 1;
    const float* src = (col < DIM) ? feat + (size_t)n * DIM + col
                                   : aggr + (size_t)n * DIM + (col - DIM);
    async_copy_b128(&X[row * XS_N + col], src);
  }
  async_wait0();
  __syncthreads();

  const int wave = t >> 5, lane = t & 31;
  const int nc   = wave * 16 + (lane & 15);

  v8f acc[RT];
  gemm_tiles_f32<2 * DIM>(X, XS_N, W1, lane, nc, b1[nc], acc);
  store_tiles(Ha, HS, lane, nc, acc, true);
  __syncthreads();
  gemm_tiles_f32<DIM>(Ha, HS, W2, lane, nc, b2[nc], acc);
  store_tiles(Hb, HS, lane, nc, acc, true);
  __syncthreads();
  gemm_tiles_f32<DIM>(Hb, HS, W3, lane, nc, b3[nc], acc);
  store_tiles(Ha, HS, lane, nc, acc, false);
  __syncthreads();

#pragma unroll
  for (int pass = 0; pass < MT / 16; ++pass) {
    const int row = (t >> 4) + pass * 16;
    const int sub = t & 15;
    float s = 0.f, ss = 0.f;
#pragma unroll
    for (int j = 0; j < 8; ++j) {
      float v = Ha[row * HS + sub * 8 + j];
      s += v; ss += v * v;
    }
#pragma unroll
    for (int off = 1; off < 16; off <<= 1) {
      s  += __shfl_xor(s,  off, 16);
      ss += __shfl_xor(ss, off, 16);
    }
    const float mu  = s * (1.f / DIM);
    const float inv = rsqrtf(ss * (1.f / DIM) - mu * mu + LN_EPS);
    const int n = r0 + row;
    if (n < N) {
#pragma unroll
      for (int j = 0; j < 8; ++j) {
        int c = sub * 8 + j;
        nout[(size_t)n * DIM + c] =
            (Ha[row * HS + c] - mu) * inv * gamma[c] + beta[c];
      }
    }
  }
}

// ---------------------------------------------------------------------------
extern "C" void kernel_launch(void* const* d_in, const int* in_sizes, int n_in,
                              void* d_out, int out_size, void* d_ws, size_t ws_size,
                              hipStream_t stream)
{
  const float* feat  = (const float*)d_in[0];
  const int*   eidx  = (const int*)  d_in[1];
  const float* eattr = (const float*)d_in[2];
  const float* eW1 = (const float*)d_in[3];  const float* eb1 = (const float*)d_in[4];
  const float* eW2 = (const float*)d_in[5];  const float* eb2 = (const float*)d_in[6];
  const float* eW3 = (const float*)d_in[7];  const float* eb3 = (const float*)d_in[8];
  const float* egm = (const float*)d_in[9];  const float* ebt = (const float*)d_in[10];
  const float* nW1 = (const float*)d_in[11]; const float* nb1 = (const float*)d_in[12];
  const float* nW2 = (const float*)d_in[13]; const float* nb2 = (const float*)d_in[14];
  const float* nW3 = (const float*)d_in[15]; const float* nb3 = (const float*)d_in[16];
  const float* ngm = (const float*)d_in[17]; const float* nbt = (const float*)d_in[18];

  const int N = in_sizes[0] / DIM;
  const int E = in_sizes[2] / DIM;

  float* out_node = (float*)d_out;                 // [N,128] first in tuple
  float* out_edge = out_node + (size_t)N * DIM;    // [E,128] second
  float* aggr     = (float*)d_ws;                  // [N,128] scatter-add accum

  const size_t smem_e =
      (size_t)(MT * XS_E + 2 * MT * HS) * sizeof(float) + 2 * MT * sizeof(int);
  const size_t smem_n = (size_t)(MT * XS_N + 2 * MT * HS) * sizeof(float);

  zero_f32<<<1024, 256, 0, stream>>>(aggr, N * DIM);
  edge_mlp<<<(E + MT - 1) / MT, 256, smem_e, stream>>>(
      feat, eidx, eattr, eW1, eb1, eW2, eb2, eW3, eb3, egm, ebt,
      out_edge, aggr, E);
  node_mlp<<<(N + MT - 1) / MT, 256, smem_n, stream>>>(
      feat, aggr, nW1, nb1, nW2, nb2, nW3, nb3, ngm, nbt,
      out_node, N);
}